// SpatialAttentionModule_1434519077055
// MI455X (gfx1250) — compile-verified
//
#include <hip/hip_runtime.h>
#include <hip/hip_bf16.h>
#include <math.h>

typedef __attribute__((ext_vector_type(2))) float v2f;
typedef __attribute__((ext_vector_type(4))) float v4f;
typedef __attribute__((ext_vector_type(8))) float v8f;

#define PLANE 16384   // 128*128
#define CH    256
#define NB    32
#define W     128
#define H     128

// ---------------------------------------------------------------------------
// Kernel 1: channel mean+max reduction.  512 MB streamed once (non-temporal),
// float4 per lane -> 512B fully-coalesced wave loads. Writes avg & max planes
// (2 MB each) to workspace; those stay resident in the 192 MB L2 for kernel 2.
// ---------------------------------------------------------------------------
__global__ __launch_bounds__(256) void reduce_mean_max_kernel(
    const float* __restrict__ x, float* __restrict__ avg, float* __restrict__ mx) {
  int tid = blockIdx.x * blockDim.x + threadIdx.x;   // 0 .. 131071
  int b   = tid >> 12;                               // 4096 float4-threads per batch
  int p4  = (tid & 4095) << 2;                       // position (multiple of 4)

  const v4f* px = (const v4f*)(x + (size_t)b * (size_t)CH * PLANE + p4);
  const int cstride4 = PLANE / 4;                    // float4 stride between channels

  v4f s = (v4f){0.f, 0.f, 0.f, 0.f};
  v4f m = (v4f){-INFINITY, -INFINITY, -INFINITY, -INFINITY};

#pragma unroll 8
  for (int c = 0; c < CH; ++c) {
    v4f v = __builtin_nontemporal_load(&px[c * cstride4]);   // TH=NT: single-use stream
    s += v;
    m.x = fmaxf(m.x, v.x);
    m.y = fmaxf(m.y, v.y);
    m.z = fmaxf(m.z, v.z);
    m.w = fmaxf(m.w, v.w);
  }

  size_t o = (size_t)b * PLANE + p4;
  const float inv = 1.0f / (float)CH;
  *(v4f*)(avg + o) = s * inv;
  *(v4f*)(mx  + o) = m;
}

// ---------------------------------------------------------------------------
// Kernel 2: 3x3 conv (2ch -> 1ch, zero pad) + SiLU via V_WMMA_F32_16X16X4_F32.
// out[n] = sum_k wt[k] * im2col[k][n], K = 18 taps padded to 20 = 5 chunks of 4.
// A[m][k] = wt[k] (rows replicated) ; B[k][n] = tap-k value at output n.
// One wave per (b,h) row; 8 tiles of 16 outputs; 5 WMMAs per tile.
// Tap decode is fully compile-time; loads are branchless (clamped address +
// value mask) so EXEC stays uniform across the whole WMMA loop.
// ---------------------------------------------------------------------------
struct Tap { int pl; int dy; int dx; int valid; };

__device__ __forceinline__ constexpr Tap tap_of(int k) {
  // k in [0,18): plane = k/9 (0=avg,1=max), dy = (k%9)/3 - 1, dx = (k%9)%3 - 1
  return Tap{ (k < 18) ? (k / 9) : 0,
              (k < 18) ? ((k % 9) / 3 - 1) : 0,
              (k < 18) ? ((k % 9) % 3 - 1) : 0,
              (k < 18) ? 1 : 0 };
}

__global__ __launch_bounds__(256) void conv3x3_silu_wmma_kernel(
    const float* __restrict__ avg, const float* __restrict__ mx,
    const float* __restrict__ cw, float* __restrict__ out) {
  int lane  = threadIdx.x & 31;
  int wave  = threadIdx.x >> 5;
  int row   = blockIdx.x * (blockDim.x >> 5) + wave;   // 0 .. 4095
  int b     = row >> 7;
  int h     = row & 127;
  int half  = lane >> 4;                               // K-pair select (0:K0/K1, 1:K2/K3)
  int n     = lane & 15;                               // output column within tile

  float wt[20];
#pragma unroll
  for (int k = 0; k < 18; ++k) wt[k] = cw[k];
  wt[18] = 0.f; wt[19] = 0.f;

  const float* p0 = avg + (size_t)b * PLANE;
  const float* p1 = mx  + (size_t)b * PLANE;

  for (int t = 0; t < 8; ++t) {
    int w0 = t * 16;
    v8f acc = {};
#pragma unroll
    for (int c = 0; c < 5; ++c) {
      // A 16x4 f32 layout: lanes 0-15 hold K=0(v0),K=1(v1); lanes 16-31 K=2,K=3
      v2f a;
      a.x = half ? wt[4 * c + 2] : wt[4 * c + 0];
      a.y = half ? wt[4 * c + 3] : wt[4 * c + 1];

      // B 4x16 f32 layout: lane=N; lanes 0-15 rows K=0(v0),K=1(v1); 16-31 K=2,K=3
      v2f bv;
#pragma unroll
      for (int s = 0; s < 2; ++s) {
        constexpr int dummy = 0; (void)dummy;
        const Tap t0 = tap_of(4 * c + s);       // half == 0 tap (compile-time)
        const Tap t1 = tap_of(4 * c + 2 + s);   // half == 1 tap (compile-time)
        int pl    = half ? t1.pl    : t0.pl;
        int dy    = half ? t1.dy    : t0.dy;
        int dx    = half ? t1.dx    : t0.dx;
        int valid = half ? t1.valid : t0.valid;

        int hh = h + dy;
        int wx = w0 + n + dx;
        bool ok = valid && (hh >= 0) && (hh < H) && (wx >= 0) && (wx < W);
        int hc = min(max(hh, 0), H - 1);        // clamp -> address always in bounds
        int wc = min(max(wx, 0), W - 1);
        const float* p = pl ? p1 : p0;
        float v = p[hc * W + wc];               // branchless load (uniform EXEC)
        v = ok ? v : 0.f;                       // zero padding / K-pad mask
        if (s == 0) bv.x = v; else bv.y = v;
      }

      acc = __builtin_amdgcn_wmma_f32_16x16x4_f32(
          /*neg_a=*/false, a, /*neg_b=*/false, bv,
          /*c_mod=*/(short)0, acc, /*reuse_a=*/false, /*reuse_b=*/false);
    }
    // D layout: v0 lanes 0-15 = row M=0 -> conv output for columns n=0..15
    if (half == 0) {
      float d  = acc[0];
      float sg = 1.0f / (1.0f + expf(-d));
      out[(size_t)b * PLANE + h * W + w0 + n] = d * sg;
    }
  }
}

extern "C" void kernel_launch(void* const* d_in, const int* in_sizes, int n_in,
                              void* d_out, int out_size, void* d_ws, size_t ws_size,
                              hipStream_t stream) {
  const float* x  = (const float*)d_in[0];   // [32,256,128,128] f32
  const float* cw = (const float*)d_in[1];   // [1,2,3,3] f32 (18 floats)
  float* out = (float*)d_out;                // [32,1,128,128] f32
  float* ws  = (float*)d_ws;                 // need 4 MB: avg plane + max plane
  float* avg = ws;
  float* mx  = ws + (size_t)NB * PLANE;

  // Kernel 1: 131072 threads (one per 4 outputs), 512 blocks x 256
  reduce_mean_max_kernel<<<512, 256, 0, stream>>>(x, avg, mx);

  // Kernel 2: 4096 rows, one wave per row, 8 waves per block -> 512 blocks
  conv3x3_silu_wmma_kernel<<<512, 256, 0, stream>>>(avg, mx, cw, out);
}